// HashEncoding_56160992363129
// MI455X (gfx1250) — compile-verified
//
#include <hip/hip_runtime.h>

// Multiresolution hash-grid encoding for gfx1250 (MI455X).
// One wave32 thread per (point, level). 8 independent global_load_b64 gathers
// per thread from the 4MB (L2-resident) table; NT temporal hints on the
// streamed position loads and feature stores so the 140MB stream does not
// evict the gather-hot table from WGP$/L2.

typedef float v2f __attribute__((ext_vector_type(2)));

#define NUM_LEVELS 16
#define PRIME_Y 2654435761u
#define PRIME_Z 805459861u
#define HASH_MASK 0x7FFFFu   // 2^19 - 1

// res_l = min(int(16 * (128)^(l/15)), 2048); stored as res-1 for clamp+scale.
__device__ __constant__ unsigned int c_rm1[NUM_LEVELS] = {
    15u,   21u,   29u,   41u,   57u,   79u,  110u,  152u,
   211u,  293u,  405u,  560u,  775u, 1071u, 1481u, 2047u
};

__global__ __launch_bounds__(256) void hashenc_kernel(
    const float* __restrict__ pos,    // (N, 3) f32 in [-1, 1]
    const v2f*   __restrict__ tbl,    // (2^19, 2) f32
    float*       __restrict__ out)    // (N, 32) f32
{
    const unsigned int tid = blockIdx.x * 256u + threadIdx.x;
    const unsigned int lvl = tid & 15u;
    const unsigned int n   = tid >> 4;

    const unsigned int rm1 = c_rm1[lvl];
    const float scale = (float)rm1;

    // 16 lanes of a wave share one point; NT: positions are read-once stream.
    float px = __builtin_nontemporal_load(pos + 3u * n + 0u);
    float py = __builtin_nontemporal_load(pos + 3u * n + 1u);
    float pz = __builtin_nontemporal_load(pos + 3u * n + 2u);
    px = (px + 1.0f) * 0.5f;                 // [-1,1] -> [0,1]
    py = (py + 1.0f) * 0.5f;
    pz = (pz + 1.0f) * 0.5f;

    const float spx = px * scale, spy = py * scale, spz = pz * scale;
    const float gx = floorf(spx), gy = floorf(spy), gz = floorf(spz);
    const float wx = spx - gx,   wy = spy - gy,   wz = spz - gz;

    unsigned int x0 = (unsigned int)gx;
    unsigned int y0 = (unsigned int)gy;
    unsigned int z0 = (unsigned int)gz;
    x0 = x0 > rm1 ? rm1 : x0;                // jnp.clip upper bound
    y0 = y0 > rm1 ? rm1 : y0;
    z0 = z0 > rm1 ? rm1 : z0;
    unsigned int x1 = x0 + 1u; x1 = x1 > rm1 ? rm1 : x1;
    unsigned int y1 = y0 + 1u; y1 = y1 > rm1 ? rm1 : y1;
    unsigned int z1 = z0 + 1u; z1 = z1 > rm1 ? rm1 : z1;

    // Hash in uint32: low 19 bits identical to the int64 reference.
    const unsigned int hy0 = y0 * PRIME_Y, hy1 = y1 * PRIME_Y;
    const unsigned int hz0 = z0 * PRIME_Z, hz1 = z1 * PRIME_Z;
    const unsigned int i000 = (x0 ^ hy0 ^ hz0) & HASH_MASK;
    const unsigned int i001 = (x0 ^ hy0 ^ hz1) & HASH_MASK;
    const unsigned int i010 = (x0 ^ hy1 ^ hz0) & HASH_MASK;
    const unsigned int i011 = (x0 ^ hy1 ^ hz1) & HASH_MASK;
    const unsigned int i100 = (x1 ^ hy0 ^ hz0) & HASH_MASK;
    const unsigned int i101 = (x1 ^ hy0 ^ hz1) & HASH_MASK;
    const unsigned int i110 = (x1 ^ hy1 ^ hz0) & HASH_MASK;
    const unsigned int i111 = (x1 ^ hy1 ^ hz1) & HASH_MASK;

    // Issue all 8 gathers before consuming: 8 global_load_b64 in flight.
    const v2f c000 = tbl[i000];
    const v2f c001 = tbl[i001];
    const v2f c010 = tbl[i010];
    const v2f c011 = tbl[i011];
    const v2f c100 = tbl[i100];
    const v2f c101 = tbl[i101];
    const v2f c110 = tbl[i110];
    const v2f c111 = tbl[i111];

    // Trilinear weights; corner c = i*4 + j*2 + k with (i,j,k) = (x,y,z) bits.
    const float ux = 1.0f - wx, uy = 1.0f - wy, uz = 1.0f - wz;
    const float a = ux * uy;   // i=0, j=0
    const float b = ux * wy;   // i=0, j=1
    const float c = wx * uy;   // i=1, j=0
    const float d = wx * wy;   // i=1, j=1
    const float t000 = a * uz, t001 = a * wz;
    const float t010 = b * uz, t011 = b * wz;
    const float t100 = c * uz, t101 = c * wz;
    const float t110 = d * uz, t111 = d * wz;

    float f0 = t000 * c000.x;
    float f1 = t000 * c000.y;
    f0 = fmaf(t001, c001.x, f0);  f1 = fmaf(t001, c001.y, f1);
    f0 = fmaf(t010, c010.x, f0);  f1 = fmaf(t010, c010.y, f1);
    f0 = fmaf(t011, c011.x, f0);  f1 = fmaf(t011, c011.y, f1);
    f0 = fmaf(t100, c100.x, f0);  f1 = fmaf(t100, c100.y, f1);
    f0 = fmaf(t101, c101.x, f0);  f1 = fmaf(t101, c101.y, f1);
    f0 = fmaf(t110, c110.x, f0);  f1 = fmaf(t110, c110.y, f1);
    f0 = fmaf(t111, c111.x, f0);  f1 = fmaf(t111, c111.y, f1);

    // out[n*32 + lvl*2 .. +1] == out[tid*2 .. +1]; coalesced across the wave.
    v2f o; o.x = f0; o.y = f1;
    __builtin_nontemporal_store(o, (v2f*)(out) + tid);
}

extern "C" void kernel_launch(void* const* d_in, const int* in_sizes, int n_in,
                              void* d_out, int out_size, void* d_ws, size_t ws_size,
                              hipStream_t stream) {
    const float* positions = (const float*)d_in[0];     // (N, 3) f32
    const v2f*   table     = (const v2f*)d_in[1];       // (2^19, 2) f32
    float*       out       = (float*)d_out;             // (N, 32) f32

    const int npts  = in_sizes[0] / 3;
    const int total = npts * NUM_LEVELS;                // one thread per (point, level)
    const int block = 256;
    const int grid  = (total + block - 1) / block;

    hashenc_kernel<<<grid, block, 0, stream>>>(positions, table, out);
}